// MyModel_61933428411126
// MI455X (gfx1250) — compile-verified
//
#include <hip/hip_runtime.h>

// FWHT of N = 2^26 fp32 values in 3 fused passes (1.5 GB traffic ~ 66us @ 23.3 TB/s).
// H16 levels run on the matrix pipe via V_WMMA_F32_16X16X4_F32 (exact fp32 math).
// Global->LDS fills use GLOBAL_LOAD_ASYNC_TO_LDS_B128 when the builtin is available.

typedef __attribute__((ext_vector_type(2))) float v2f;
typedef __attribute__((ext_vector_type(8))) float v8f;
typedef __attribute__((ext_vector_type(4))) int   v4i;

#if __has_builtin(__builtin_amdgcn_global_load_async_to_lds_b128) && \
    __has_builtin(__builtin_amdgcn_s_wait_asynccnt)
#define FWHT_ASYNC_LDS 1
typedef __attribute__((address_space(1))) v4i g_v4i;   // global int4
typedef __attribute__((address_space(3))) v4i l_v4i;   // LDS int4
#else
#define FWHT_ASYNC_LDS 0
#endif

__device__ __forceinline__ void fill16(float* ldsDst, const float* gSrc) {
#if FWHT_ASYNC_LDS
  __builtin_amdgcn_global_load_async_to_lds_b128(
      (g_v4i*)gSrc, (l_v4i*)ldsDst, /*offset=*/0, /*cpol=*/0);
#else
  *(float4*)ldsDst = *(const float4*)gSrc;
#endif
}

__device__ __forceinline__ void fill_fence() {
#if FWHT_ASYNC_LDS
  __builtin_amdgcn_s_wait_asynccnt(0);
#endif
}

// Apply H16 along the 4 tile-index bits at position P of an E-float LDS tile.
// Y[m][n] = sum_k H16[m][k] * X[k][n],  H16[m][k] = (-1)^popcount(m&k)
// element(t, c) = lds[ base(c) + t*S ],  S = 1<<P, 16 columns n per WMMA tile.
template<int E, int P>
__device__ __forceinline__ void wmma_h16_level(float* lds) {
  constexpr int S     = 1 << P;
  constexpr int NT    = (E / 16) / 16;   // 16x16 WMMA tiles in this level
  constexpr int ITERS = NT / 8;          // tiles per wave (8 waves/block), uniform
  const int tid  = threadIdx.x;
  const int lane = tid & 31;
  const int l16  = lane & 15;
  const int half = lane >> 4;            // wave32: lanes 0-15 / 16-31
  const int wave = tid >> 5;             // 8 waves per 256-thread block

  // A fragment (16x4 f32, 2 VGPRs): lane L holds A[L%16][2*(L/16)+v]
  v2f a[4];
#pragma unroll
  for (int kb = 0; kb < 4; ++kb) {
#pragma unroll
    for (int v = 0; v < 2; ++v) {
      const int k = 4 * kb + 2 * half + v;
      a[kb][v] = (__builtin_popcount(l16 & k) & 1) ? -1.0f : 1.0f;
    }
  }

  // Compile-time trip count: no divergent loop structure, EXEC stays all-ones.
#pragma unroll
  for (int it = 0; it < ITERS; ++it) {
    const int T    = it * 8 + wave;
    const int c    = T * 16 + l16;                       // this lane's column id
    const int base = ((c >> P) << (P + 4)) | (c & (S - 1));
    v8f acc = {};
#pragma unroll
    for (int kb = 0; kb < 4; ++kb) {
      // B fragment (4x16 f32): lane L holds B[2*(L/16)+v][L%16]
      v2f b;
      b[0] = lds[base + (4 * kb + 2 * half + 0) * S];
      b[1] = lds[base + (4 * kb + 2 * half + 1) * S];
      acc = __builtin_amdgcn_wmma_f32_16x16x4_f32(
          /*neg_a=*/false, a[kb], /*neg_b=*/false, b,
          /*c_mod=*/(short)0, acc, /*reuse_a=*/false, /*reuse_b=*/false);
    }
    // D fragment: lane L, VGPR v -> D[v + 8*(L/16)][L%16]
#pragma unroll
    for (int v = 0; v < 8; ++v)
      lds[base + (v + 8 * half) * S] = acc[v];
  }
}

// ---------- Pass 1: bits [0,12), contiguous 4096-float tiles ----------
__global__ __launch_bounds__(256) void fwht_pass1(const float* __restrict__ in,
                                                  float* __restrict__ out) {
  __shared__ __align__(16) float lds[4096];
  const int tid = threadIdx.x;
  const size_t blockBase = (size_t)blockIdx.x * 4096;

#pragma unroll
  for (int it = 0; it < 4; ++it) {
    const int i = it * 256 + tid;
    fill16(lds + i * 4, in + blockBase + i * 4);
  }
  fill_fence();
  __syncthreads();

  wmma_h16_level<4096, 0>(lds);  __syncthreads();   // bits [0,4)
  wmma_h16_level<4096, 4>(lds);  __syncthreads();   // bits [4,8)
  wmma_h16_level<4096, 8>(lds);  __syncthreads();   // bits [8,12)

#pragma unroll
  for (int it = 0; it < 4; ++it) {
    const int i = it * 256 + tid;
    ((float4*)(out + blockBase))[i] = ((float4*)lds)[i];
  }
}

// ---------- Pass 2: bits [12,20), tile = 256 rows (stride 2^12) x 32 floats ----------
__global__ __launch_bounds__(256) void fwht_pass2(float* __restrict__ data) {
  __shared__ __align__(16) float lds[8192];
  const int tid = threadIdx.x;
  const int bid = blockIdx.x;                 // 8192 blocks
  const int f1  = bid & 127;                  // global bits [5,12)
  const int f2  = bid >> 7;                   // global bits [20,26)
  const size_t base = ((size_t)f2 << 20) | ((size_t)f1 << 5);

#pragma unroll
  for (int it = 0; it < 8; ++it) {            // 128B contiguous per row, coalesced
    const int i = it * 256 + tid;
    const int r = i >> 3, q = i & 7;
    fill16(lds + i * 4, data + base + ((size_t)r << 12) + q * 4);
  }
  fill_fence();
  __syncthreads();

  wmma_h16_level<8192, 5>(lds);  __syncthreads();   // row bits [0,4) = global [12,16)
  wmma_h16_level<8192, 9>(lds);  __syncthreads();   // row bits [4,8) = global [16,20)

#pragma unroll
  for (int it = 0; it < 8; ++it) {
    const int i = it * 256 + tid;
    const int r = i >> 3, q = i & 7;
    ((float4*)(data + base + ((size_t)r << 12)))[q] = ((float4*)lds)[i];
  }
}

// ---------- Pass 3: bits [20,26), tile = 64 rows (stride 2^20) x 128 floats ----------
__global__ __launch_bounds__(256) void fwht_pass3(float* __restrict__ data) {
  __shared__ __align__(16) float lds[8192];
  const int tid = threadIdx.x;
  const size_t base = (size_t)blockIdx.x << 7;     // global bits [7,20)

#pragma unroll
  for (int it = 0; it < 8; ++it) {                 // 512B contiguous per row
    const int i = it * 256 + tid;
    const int r = i >> 5, q = i & 31;
    fill16(lds + i * 4, data + base + ((size_t)r << 20) + q * 4);
  }
  fill_fence();
  __syncthreads();

  wmma_h16_level<8192, 7>(lds);  __syncthreads();  // row bits [0,4) = global [20,24)

  // H4 on tile-index bits 11,12 (global bits 24,25): plain VALU butterflies
#pragma unroll
  for (int lb = 11; lb <= 12; ++lb) {
    const int st = 1 << lb;
#pragma unroll
    for (int it = 0; it < 16; ++it) {
      const int q   = it * 256 + tid;
      const int idx = ((q >> lb) << (lb + 1)) | (q & (st - 1));
      const float u = lds[idx], w = lds[idx + st];
      lds[idx]      = u + w;
      lds[idx + st] = u - w;
    }
    __syncthreads();
  }

#pragma unroll
  for (int it = 0; it < 8; ++it) {
    const int i = it * 256 + tid;
    const int r = i >> 5, q = i & 31;
    ((float4*)(data + base + ((size_t)r << 20)))[q] = ((float4*)lds)[i];
  }
}

extern "C" void kernel_launch(void* const* d_in, const int* in_sizes, int n_in,
                              void* d_out, int out_size, void* d_ws, size_t ws_size,
                              hipStream_t stream) {
  (void)n_in; (void)d_ws; (void)ws_size; (void)out_size;
  const float* x   = (const float*)d_in[0];
  float*       out = (float*)d_out;
  const unsigned n = (unsigned)in_sizes[0];        // 2^26

  dim3 blk(256);
  fwht_pass1<<<dim3(n / 4096), blk, 0, stream>>>(x, out);   // bits [0,12)
  fwht_pass2<<<dim3(n / 8192), blk, 0, stream>>>(out);      // bits [12,20)
  fwht_pass3<<<dim3(n / 8192), blk, 0, stream>>>(out);      // bits [20,26)
}